// AttentionAggregator_24893630447803
// MI455X (gfx1250) — compile-verified
//
#include <hip/hip_runtime.h>

typedef float v2f __attribute__((ext_vector_type(2)));
typedef float v8f __attribute__((ext_vector_type(8)));
typedef int   v4i __attribute__((ext_vector_type(4)));

#define DF 128          // feature dim
#define LP 132          // padded LDS row (floats) -> conflict-free frag reads

#if defined(__has_builtin)
#if __has_builtin(__builtin_amdgcn_global_load_async_to_lds_b128)
#define USE_ASYNC_LDS 1
#endif
#endif

#ifdef USE_ASYNC_LDS
typedef __attribute__((address_space(1))) v4i gv4i;  // global int4
typedef __attribute__((address_space(3))) v4i lv4i;  // LDS int4
#endif

__device__ __forceinline__ float lrelu(float v) { return v > 0.f ? v : 0.2f * v; }

__device__ __forceinline__ float wave_sum32(float v) {
#pragma unroll
  for (int off = 16; off >= 1; off >>= 1) v += __shfl_xor(v, off, 32);
  return v;
}

// ---------------------------------------------------------------------------
// Kernel A: persistent WMMA GEMM. Per 16-node tile computes
//   h0 = relu(x W0^T + b0), h1 = relu(x W1^T + b1)
// and fuses: a_self = lrelu(h0 . att[:128]), a_neigh = lrelu(h1 . att[128:]),
//   out = norm(h0, scale0, offset0), h1 -> global (for edge gather).
// Block = 256 threads = 8 waves; wave w owns output-feature tile [16w,16w+16).
// W fragments are register-resident for the whole kernel. x tiles staged to
// LDS with CDNA5 async-to-LDS loads (ASYNCcnt) when available.
// ---------------------------------------------------------------------------
__global__ __launch_bounds__(256)
void gat_gemm_kernel(const float* __restrict__ x,
                     const float* __restrict__ W0, const float* __restrict__ b0,
                     const float* __restrict__ W1, const float* __restrict__ b1,
                     const float* __restrict__ att,
                     const float* __restrict__ scale0, const float* __restrict__ offset0,
                     float* __restrict__ out, float* __restrict__ h1g,
                     float* __restrict__ a_self, float* __restrict__ a_neigh,
                     int nTiles) {
  __shared__ float ldsX[16][LP];
  __shared__ float ldsH[16][LP];

  const int tid  = threadIdx.x;
  const int lane = tid & 31;
  const int wave = tid >> 5;
  const int hi   = lane >> 4;   // half-wave: selects K pair (A/B) / M+8 (C)
  const int mn   = lane & 15;   // M for A-frag, N for B/C/D frags
  const int n0   = wave * 16;   // output-feature tile base

  // Register-resident W fragments: lane needs W[n0+mn][4kt + 2hi + {0,1}]
  v2f w0f[32], w1f[32];
  const float* w0row = W0 + (size_t)(n0 + mn) * DF + 2 * hi;
  const float* w1row = W1 + (size_t)(n0 + mn) * DF + 2 * hi;
#pragma unroll
  for (int kt = 0; kt < 32; ++kt) {
    w0f[kt] = *(const v2f*)(w0row + 4 * kt);
    w1f[kt] = *(const v2f*)(w1row + 4 * kt);
  }
  const float b0v = b0[n0 + mn];
  const float b1v = b1[n0 + mn];

  // per-lane epilogue params (lane covers features lane*4 .. lane*4+3)
  const float4 sc0   = *(const float4*)(scale0  + lane * 4);
  const float4 of0   = *(const float4*)(offset0 + lane * 4);
  const float4 attLo = *(const float4*)(att       + lane * 4);
  const float4 attHi = *(const float4*)(att + DF  + lane * 4);

  for (int tile = blockIdx.x; tile < nTiles; tile += gridDim.x) {
    const int m0 = tile * 16;

    // (A) coalesced load of 16x128 x-tile into LDS
#pragma unroll
    for (int idx = tid; idx < 16 * 32; idx += 256) {
      const int r  = idx >> 5;
      const int c4 = (idx & 31) * 4;
      const float* src = x + (size_t)(m0 + r) * DF + c4;
#ifdef USE_ASYNC_LDS
      __builtin_amdgcn_global_load_async_to_lds_b128(
          (gv4i*)src, (lv4i*)&ldsX[r][c4], 0, 0);
#else
      *(float4*)&ldsX[r][c4] = *(const float4*)src;
#endif
    }
#ifdef USE_ASYNC_LDS
#if __has_builtin(__builtin_amdgcn_s_wait_asynccnt)
    __builtin_amdgcn_s_wait_asynccnt(0);
#else
    asm volatile("s_wait_asynccnt 0x0" ::: "memory");
#endif
#endif
    __syncthreads();

    // (B) K-loop: 32 x (V_WMMA_F32_16X16X4_F32 x2), A frags from LDS
    v8f acc0 = {}, acc1 = {};
#pragma unroll
    for (int kt = 0; kt < 32; ++kt) {
      const v2f a = *(const v2f*)&ldsX[mn][4 * kt + 2 * hi];
      acc0 = __builtin_amdgcn_wmma_f32_16x16x4_f32(false, a, false, w0f[kt],
                                                   (short)0, acc0, false, false);
      acc1 = __builtin_amdgcn_wmma_f32_16x16x4_f32(false, a, false, w1f[kt],
                                                   (short)0, acc1, false, false);
    }

    // bias + relu
    float h0v[8], h1v[8];
#pragma unroll
    for (int r = 0; r < 8; ++r) {
      h0v[r] = fmaxf(acc0[r] + b0v, 0.f);
      h1v[r] = fmaxf(acc1[r] + b1v, 0.f);
    }

    // (C) h0 fragments -> LDS tile (D-frag: row = r + 8*hi, col = n0+mn)
#pragma unroll
    for (int r = 0; r < 8; ++r) ldsH[r + 8 * hi][n0 + mn] = h0v[r];
    __syncthreads();

    // (D) per wave: 2 nodes — norm(h0) -> out, a_self
#pragma unroll
    for (int t = 0; t < 2; ++t) {
      const int node = wave * 2 + t;
      const float4 v = *(const float4*)&ldsH[node][lane * 4];
      float s  = v.x + v.y + v.z + v.w;
      float sq = v.x * v.x + v.y * v.y + v.z * v.z + v.w * v.w;
      float ad = v.x * attLo.x + v.y * attLo.y + v.z * attLo.z + v.w * attLo.w;
      s = wave_sum32(s); sq = wave_sum32(sq); ad = wave_sum32(ad);
      const float mean = s * (1.f / 128.f);
      const float var  = sq * (1.f / 128.f) - mean * mean;
      const float rinv = rsqrtf(var + 1e-9f);
      float4 o;
      o.x = (v.x - mean) * sc0.x * rinv + of0.x;
      o.y = (v.y - mean) * sc0.y * rinv + of0.y;
      o.z = (v.z - mean) * sc0.z * rinv + of0.z;
      o.w = (v.w - mean) * sc0.w * rinv + of0.w;
      *(float4*)(out + (size_t)(m0 + node) * DF + lane * 4) = o;
      if (lane == 0) a_self[m0 + node] = lrelu(ad);
    }
    __syncthreads();

    // (E) h1 fragments -> LDS tile
#pragma unroll
    for (int r = 0; r < 8; ++r) ldsH[r + 8 * hi][n0 + mn] = h1v[r];
    __syncthreads();

    // (F) per wave: 2 nodes — a_neigh + coalesced h1 store
#pragma unroll
    for (int t = 0; t < 2; ++t) {
      const int node = wave * 2 + t;
      const float4 v = *(const float4*)&ldsH[node][lane * 4];
      float ad = v.x * attHi.x + v.y * attHi.y + v.z * attHi.z + v.w * attHi.w;
      ad = wave_sum32(ad);
      *(float4*)(h1g + (size_t)(m0 + node) * DF + lane * 4) = v;
      if (lane == 0) a_neigh[m0 + node] = lrelu(ad);
    }
    // next iteration's (A)+barrier protects ldsH/ldsX reuse
  }
}

// ---------------------------------------------------------------------------
// Edge binning: histogram -> single-block scan -> scatter of col[]
// ---------------------------------------------------------------------------
__global__ void hist_kernel(const int* __restrict__ row, unsigned* __restrict__ counts, int E) {
  const int i = blockIdx.x * blockDim.x + threadIdx.x;
  if (i < E) atomicAdd(&counts[row[i]], 1u);
}

__global__ __launch_bounds__(1024)
void scan_kernel(const unsigned* __restrict__ counts, unsigned* __restrict__ offsets, int n) {
  __shared__ unsigned sums[1024];
  const int tid   = threadIdx.x;
  const int chunk = (n + 1023) / 1024;
  const int beg   = tid * chunk;
  const int end   = min(beg + chunk, n);
  unsigned s = 0;
  for (int i = beg; i < end; ++i) s += counts[i];
  sums[tid] = s;
  __syncthreads();
  for (int off = 1; off < 1024; off <<= 1) {   // Hillis-Steele inclusive scan
    unsigned v = (tid >= off) ? sums[tid - off] : 0u;
    __syncthreads();
    sums[tid] += v;
    __syncthreads();
  }
  unsigned run = (tid == 0) ? 0u : sums[tid - 1];
  for (int i = beg; i < end; ++i) { offsets[i] = run; run += counts[i]; }
  if (beg < n && end == n) offsets[n] = run;   // total
}

__global__ void scatter_kernel(const int* __restrict__ row, const int* __restrict__ col,
                               unsigned* __restrict__ cursor, int* __restrict__ permcol, int E) {
  const int i = blockIdx.x * blockDim.x + threadIdx.x;
  if (i < E) {
    const unsigned pos = atomicAdd(&cursor[row[i]], 1u);
    permcol[pos] = col[i];
  }
}

// ---------------------------------------------------------------------------
// Kernel E: one wave per destination node. Register accumulation of
//   agg[i] = sum_j (a_self[i] + a_neigh[j]) * h1[j]   (h1 rows hit L2)
// then fused norm(agg, scale1, offset1) added into out. No float atomics.
// Index stream rotated one iteration ahead to overlap the dependent loads.
// ---------------------------------------------------------------------------
__global__ __launch_bounds__(256)
void aggregate_kernel(const int* __restrict__ permcol, const unsigned* __restrict__ offsets,
                      const float* __restrict__ a_self, const float* __restrict__ a_neigh,
                      const float* __restrict__ h1g,
                      const float* __restrict__ scale1, const float* __restrict__ offset1,
                      float* __restrict__ out, int N) {
  const int node = (int)((blockIdx.x * (unsigned)blockDim.x + threadIdx.x) >> 5);
  const int lane = threadIdx.x & 31;
  if (node >= N) return;
  const unsigned beg = offsets[node], end = offsets[node + 1];
  const float asl = a_self[node];
  float4 acc = {0.f, 0.f, 0.f, 0.f};
  if (beg < end) {
    int j = permcol[beg];                       // head of pipeline
    for (unsigned k = beg; k < end; ++k) {
      const int jn = (k + 1 < end) ? permcol[k + 1] : 0;  // overlap next index
      const float e  = asl + a_neigh[j];
      const float4 h = *(const float4*)(h1g + (size_t)j * DF + lane * 4);
      acc.x = fmaf(e, h.x, acc.x);
      acc.y = fmaf(e, h.y, acc.y);
      acc.z = fmaf(e, h.z, acc.z);
      acc.w = fmaf(e, h.w, acc.w);
      j = jn;
    }
  }
  float s  = acc.x + acc.y + acc.z + acc.w;
  float sq = acc.x * acc.x + acc.y * acc.y + acc.z * acc.z + acc.w * acc.w;
  s = wave_sum32(s); sq = wave_sum32(sq);
  const float mean = s * (1.f / 128.f);
  const float var  = sq * (1.f / 128.f) - mean * mean;
  const float rinv = rsqrtf(var + 1e-9f);
  const float4 sc = *(const float4*)(scale1  + lane * 4);
  const float4 of = *(const float4*)(offset1 + lane * 4);
  float* orow = out + (size_t)node * DF + lane * 4;
  float4 o = *(const float4*)orow;
  o.x += (acc.x - mean) * sc.x * rinv + of.x;
  o.y += (acc.y - mean) * sc.y * rinv + of.y;
  o.z += (acc.z - mean) * sc.z * rinv + of.z;
  o.w += (acc.w - mean) * sc.w * rinv + of.w;
  *(float4*)orow = o;
}

// ---------------------------------------------------------------------------
static inline size_t alignUp(size_t v, size_t a) { return (v + a - 1) & ~(a - 1); }

extern "C" void kernel_launch(void* const* d_in, const int* in_sizes, int n_in,
                              void* d_out, int out_size, void* d_ws, size_t ws_size,
                              hipStream_t stream) {
  const float* x       = (const float*)d_in[0];
  const int*   row     = (const int*)  d_in[1];
  const int*   col     = (const int*)  d_in[2];
  const float* W0      = (const float*)d_in[3];
  const float* b0      = (const float*)d_in[4];
  const float* W1      = (const float*)d_in[5];
  const float* b1      = (const float*)d_in[6];
  const float* att     = (const float*)d_in[7];
  const float* scale0  = (const float*)d_in[8];
  const float* offset0 = (const float*)d_in[9];
  const float* scale1  = (const float*)d_in[10];
  const float* offset1 = (const float*)d_in[11];
  float* out = (float*)d_out;

  const int N = in_sizes[0] / DF;
  const int E = in_sizes[1];

  // workspace carve-up
  uintptr_t p = alignUp((uintptr_t)d_ws, 256);
  float*    h1g     = (float*)p;              p = alignUp(p + (size_t)N * DF * 4, 256);
  float*    a_self  = (float*)p;              p = alignUp(p + (size_t)N * 4, 256);
  float*    a_neigh = (float*)p;              p = alignUp(p + (size_t)N * 4, 256);
  unsigned* counts  = (unsigned*)p;           p = alignUp(p + (size_t)N * 4, 256);
  unsigned* offsets = (unsigned*)p;           p = alignUp(p + ((size_t)N + 1) * 4, 256);
  unsigned* cursor  = (unsigned*)p;           p = alignUp(p + ((size_t)N + 1) * 4, 256);
  int*      permcol = (int*)p;

  (void)hipMemsetAsync(counts, 0, (size_t)N * 4, stream);

  const int nTiles = N / 16;
  const int gemmBlocks = nTiles < 1280 ? nTiles : 1280;
  gat_gemm_kernel<<<gemmBlocks, 256, 0, stream>>>(
      x, W0, b0, W1, b1, att, scale0, offset0, out, h1g, a_self, a_neigh, nTiles);

  const int eBlocks = (E + 255) / 256;
  hist_kernel<<<eBlocks, 256, 0, stream>>>(row, counts, E);
  scan_kernel<<<1, 1024, 0, stream>>>(counts, offsets, N);
  (void)hipMemcpyAsync(cursor, offsets, (size_t)N * 4, hipMemcpyDeviceToDevice, stream);
  scatter_kernel<<<eBlocks, 256, 0, stream>>>(row, col, cursor, permcol, E);

  const int aggBlocks = (N + 7) / 8;  // 8 waves (nodes) per 256-thread block
  aggregate_kernel<<<aggBlocks, 256, 0, stream>>>(
      permcol, offsets, a_self, a_neigh, h1g, scale1, offset1, out, N);
}